// GATv2_Layer_48112223649929
// MI455X (gfx1250) — compile-verified
//
#include <hip/hip_runtime.h>
#include <hip/hip_bf16.h>
#include <math.h>

#define N_NODES   50000
#define N_EDGES   800000
#define IN_FEATS  128
#define N_HEADS   4
#define OUT_FEATS 32
#define HF        (N_HEADS * OUT_FEATS)   // 128
#define SLOPE     0.2f

typedef float v2f __attribute__((ext_vector_type(2)));
typedef float v8f __attribute__((ext_vector_type(8)));

// ---------------------------------------------------------------------------
// Fused projection GEMM: P = feat @ W^T + b for W in {W_src, W_dst, W_res}.
// One wave computes one 16x16 output tile via 32x V_WMMA_F32_16X16X4_F32.
//   mat 0 -> feat_src (ws), mat 1 -> feat_dst (ws), mat 2 -> residual (d_out)
// ---------------------------------------------------------------------------
__global__ void gat_proj_wmma(const float* __restrict__ feat,
                              const float* __restrict__ Wsrc, const float* __restrict__ bsrc,
                              const float* __restrict__ Wdst, const float* __restrict__ bdst,
                              const float* __restrict__ Wres, const float* __restrict__ bres,
                              float* __restrict__ fs, float* __restrict__ fd,
                              float* __restrict__ out)
{
    const int NT = N_NODES / 16;        // 3125 node tiles
    const int OT = 3 * (HF / 16);       // 24 column tiles (3 matrices x 8)
    int wave = (int)((blockIdx.x * blockDim.x + threadIdx.x) >> 5);
    int lane = threadIdx.x & 31;
    if (wave >= NT * OT) return;

    int nodeTile = wave / OT;
    int ot       = wave % OT;
    int mat      = ot >> 3;             // 0=src, 1=dst, 2=res
    int colBase  = (ot & 7) * 16;

    const float* W; const float* b; float* dstp;
    if (mat == 0)      { W = Wsrc; b = bsrc; dstp = fs;  }
    else if (mat == 1) { W = Wdst; b = bdst; dstp = fd;  }
    else               { W = Wres; b = bres; dstp = out; }

    // A: row = lane&15, K = k0 + 2*(lane>>4) + {0,1}
    // B: col = lane&15, same K pattern;  B[k][n] = W[colBase+n][k]
    const int mrow  = lane & 15;
    const int half2 = (lane >> 4) * 2;
    const float* arow = feat + (size_t)(nodeTile * 16 + mrow) * IN_FEATS + half2;
    const float* brow = W    + (size_t)(colBase + mrow) * IN_FEATS + half2;

    v8f c = {};
#pragma unroll
    for (int k0 = 0; k0 < IN_FEATS; k0 += 4) {
        v2f a  = *(const v2f*)(arow + k0);   // 8-byte aligned: k0+half2 even
        v2f bb = *(const v2f*)(brow + k0);
        c = __builtin_amdgcn_wmma_f32_16x16x4_f32(
                /*neg_a=*/false, a, /*neg_b=*/false, bb,
                /*c_mod=*/(short)0, c, /*reuse_a=*/false, /*reuse_b=*/false);
    }

    // C/D: VGPR i -> row = i + 8*(lane>>4), col = lane&15
    const int ccol  = lane & 15;
    const float bias = b[colBase + ccol];
    const int rbase = nodeTile * 16 + (lane >> 4) * 8;
#pragma unroll
    for (int i = 0; i < 8; ++i) {
        dstp[(size_t)(rbase + i) * HF + colBase + ccol] = c[i] + bias;
    }
}

// ---------------------------------------------------------------------------
// Init per-(node,head) running max (-inf) and denominator (0).
// ---------------------------------------------------------------------------
__global__ void gat_init(float* __restrict__ nodemax, float* __restrict__ den, int n)
{
    int i = blockIdx.x * blockDim.x + threadIdx.x;
    if (i < n) { nodemax[i] = -INFINITY; den[i] = 0.0f; }
}

// Sign-aware float atomic max using integer atomics (monotone bit encodings).
__device__ __forceinline__ void atomicMaxFloat(float* addr, float val)
{
    if (val >= 0.0f) atomicMax((int*)addr, __float_as_int(val));
    else             atomicMin((unsigned int*)addr, __float_as_uint(val));
}

// ---------------------------------------------------------------------------
// Pass 1: wave per edge. score[e][h] = sum_f attn[h][f]*leaky(fs[src]+fd[dst]),
// plus segment max into nodemax[dst][h].
// ---------------------------------------------------------------------------
__global__ void gat_edge_score(const int* __restrict__ src, const int* __restrict__ dst,
                               const float* __restrict__ fs, const float* __restrict__ fd,
                               const float* __restrict__ attn,
                               float* __restrict__ score, float* __restrict__ nodemax)
{
    int wave = (int)((blockIdx.x * blockDim.x + threadIdx.x) >> 5);
    int lane = threadIdx.x & 31;
    if (wave >= N_EDGES) return;

    const int s = src[wave], d = dst[wave];
    const float* fsr = fs + (size_t)s * HF;
    const float* fdr = fd + (size_t)d * HF;

#pragma unroll
    for (int h = 0; h < N_HEADS; ++h) {
        float v = fsr[h * OUT_FEATS + lane] + fdr[h * OUT_FEATS + lane];
        v = v > 0.0f ? v : SLOPE * v;                 // LeakyReLU
        float p = v * attn[h * OUT_FEATS + lane];
#pragma unroll
        for (int off = 16; off > 0; off >>= 1)        // wave32 tree reduce
            p += __shfl_xor(p, off, 32);
        if (lane == 0) {
            score[(size_t)wave * N_HEADS + h] = p;
            atomicMaxFloat(&nodemax[d * N_HEADS + h], p);
        }
    }
}

// ---------------------------------------------------------------------------
// Pass 2: thread per (edge,head). ex = exp(score - m[dst]); den[dst] += ex.
// ---------------------------------------------------------------------------
__global__ void gat_edge_exp(const int* __restrict__ dst,
                             float* __restrict__ score,           // in: score, out: ex
                             const float* __restrict__ nodemax,
                             float* __restrict__ den)
{
    int i = blockIdx.x * blockDim.x + threadIdx.x;
    if (i >= N_EDGES * N_HEADS) return;
    int e = i >> 2, h = i & 3;
    int d = dst[e];
    float m = nodemax[d * N_HEADS + h];
    if (m == -INFINITY) m = 0.0f;                     // jnp.where(isfinite(m), m, 0)
    float ex = expf(score[i] - m);
    score[i] = ex;
    atomicAdd(&den[d * N_HEADS + h], ex);
}

// ---------------------------------------------------------------------------
// Pass 3: wave per edge. out[dst][h][f] += fs[src][h][f] * (ex / den[dst][h]).
// d_out already holds the residual projection from gat_proj_wmma.
// ---------------------------------------------------------------------------
__global__ void gat_aggregate(const int* __restrict__ src, const int* __restrict__ dst,
                              const float* __restrict__ fs,
                              const float* __restrict__ ex, const float* __restrict__ den,
                              float* __restrict__ out)
{
    int wave = (int)((blockIdx.x * blockDim.x + threadIdx.x) >> 5);
    int lane = threadIdx.x & 31;
    if (wave >= N_EDGES) return;

    const int s = src[wave], d = dst[wave];
    const float* fsr = fs + (size_t)s * HF;
#pragma unroll
    for (int h = 0; h < N_HEADS; ++h) {
        float a = ex[(size_t)wave * N_HEADS + h] / den[d * N_HEADS + h];
        atomicAdd(&out[(size_t)d * HF + h * OUT_FEATS + lane],
                  fsr[h * OUT_FEATS + lane] * a);
    }
}

// ---------------------------------------------------------------------------
// Launch
// ---------------------------------------------------------------------------
extern "C" void kernel_launch(void* const* d_in, const int* in_sizes, int n_in,
                              void* d_out, int out_size, void* d_ws, size_t ws_size,
                              hipStream_t stream)
{
    const float* feat  = (const float*)d_in[0];
    const float* Wsrc  = (const float*)d_in[1];
    const float* bsrc  = (const float*)d_in[2];
    const float* Wdst  = (const float*)d_in[3];
    const float* bdst  = (const float*)d_in[4];
    const float* attn  = (const float*)d_in[5];
    const float* Wres  = (const float*)d_in[6];
    const float* bres  = (const float*)d_in[7];
    const int*   src   = (const int*)d_in[8];
    const int*   dst   = (const int*)d_in[9];
    float*       out   = (float*)d_out;

    // Workspace carve-up (floats): fs | fd | score | nodemax | den
    float* fs      = (float*)d_ws;
    float* fd      = fs + (size_t)N_NODES * HF;
    float* score   = fd + (size_t)N_NODES * HF;
    float* nodemax = score + (size_t)N_EDGES * N_HEADS;
    float* den     = nodemax + (size_t)N_NODES * N_HEADS;

    // 0) init segment max / denom
    {
        int n = N_NODES * N_HEADS;
        gat_init<<<(n + 255) / 256, 256, 0, stream>>>(nodemax, den, n);
    }
    // 1) fused WMMA projections (fs, fd to ws; residual+bias to d_out)
    {
        int waves   = (N_NODES / 16) * 24;          // 75000 tiles
        int threads = waves * 32;
        gat_proj_wmma<<<(threads + 255) / 256, 256, 0, stream>>>(
            feat, Wsrc, bsrc, Wdst, bdst, Wres, bres, fs, fd, out);
    }
    // 2) edge scores + segment max
    {
        long long threads = (long long)N_EDGES * 32;
        gat_edge_score<<<(int)((threads + 255) / 256), 256, 0, stream>>>(
            src, dst, fs, fd, attn, score, nodemax);
    }
    // 3) exp + segment sum
    {
        int n = N_EDGES * N_HEADS;
        gat_edge_exp<<<(n + 255) / 256, 256, 0, stream>>>(dst, score, nodemax, den);
    }
    // 4) normalize + scatter-add aggregation into residual
    {
        long long threads = (long long)N_EDGES * 32;
        gat_aggregate<<<(int)((threads + 255) / 256), 256, 0, stream>>>(
            src, dst, fs, score, den, out);
    }
}